// GCN_pairs_distance_82806969467502
// MI455X (gfx1250) — compile-verified
//
#include <hip/hip_runtime.h>
#include <hip/hip_bf16.h>
#include <math.h>

#define HID 128
#define DOUT 64
#define N_GRAPHS 512
#define EPS_PD 1e-6f

typedef float v2f __attribute__((ext_vector_type(2)));
typedef float v8f __attribute__((ext_vector_type(8)));

static inline int ceil_div(int a, int b) { return (a + b - 1) / b; }

// ---------------------------------------------------------------- utilities
__global__ void fill_f32(float* __restrict__ p, float v, int n) {
  int i = blockIdx.x * blockDim.x + threadIdx.x;
  if (i < n) p[i] = v;
}

__global__ void degree_accum(const int* __restrict__ dst, float* __restrict__ deg, int E) {
  int e = blockIdx.x * blockDim.x + threadIdx.x;
  if (e < E) atomicAdd(deg + dst[e], 1.0f);
}

__global__ void rsqrt_inplace(float* __restrict__ deg, int n) {
  int i = blockIdx.x * blockDim.x + threadIdx.x;
  if (i < n) {
    float d = deg[i];
    deg[i] = (d > 0.0f) ? rsqrtf(d) : 0.0f;
  }
}

// WT[N,K] = W[K,N]^T so B fragments load as contiguous 8-byte pairs
__global__ void transpose_w(const float* __restrict__ W, float* __restrict__ WT, int K, int N) {
  int i = blockIdx.x * blockDim.x + threadIdx.x;
  if (i >= K * N) return;
  int k = i / N, n = i - k * N;
  WT[(size_t)n * K + k] = W[i];
}

// ------------------------------------------------- fp32 WMMA GEMM (16x16x4)
// H[M,N] = X[M,K] @ W[K,N] (+ bias), W given transposed as WT[N,K].
// Each wave computes a 16x64 strip: one A fragment feeds 4 WMMAs per k-step.
// block = (32, N/64), grid = (M/16). Requires M%16==0, N%64==0, K%4==0.
__global__ void wmma_gemm_f32(const float* __restrict__ X, const float* __restrict__ WT,
                              const float* __restrict__ bias, float* __restrict__ H,
                              int M, int N, int K) {
  const int mt   = blockIdx.x * 16;
  const int nt0  = threadIdx.y * 64;
  const int lane = threadIdx.x;        // 0..31
  const int half = lane >> 4;          // 0: K pair {k,k+1}; 1: {k+2,k+3}
  const int l    = lane & 15;          // row (A) / col (B,C)

  const float* __restrict__ xrow = X + (size_t)(mt + l) * K;
  const float* __restrict__ wt0 = WT + (size_t)(nt0 +  0 + l) * K;
  const float* __restrict__ wt1 = WT + (size_t)(nt0 + 16 + l) * K;
  const float* __restrict__ wt2 = WT + (size_t)(nt0 + 32 + l) * K;
  const float* __restrict__ wt3 = WT + (size_t)(nt0 + 48 + l) * K;

  v8f acc0 = {}, acc1 = {}, acc2 = {}, acc3 = {};
  for (int k = 0; k < K; k += 4) {
    const int ka = k + half * 2;
    // A 16x4 f32: lanes 0-15 (M=0..15) hold K={k,k+1}; lanes 16-31 hold K={k+2,k+3}
    const v2f a  = *(const v2f*)(xrow + ka);
    // B 4x16 f32 (mirror): lanes 0-15 (N) hold rows {k,k+1}; lanes 16-31 rows {k+2,k+3}
    const v2f b0 = *(const v2f*)(wt0 + ka);
    const v2f b1 = *(const v2f*)(wt1 + ka);
    const v2f b2 = *(const v2f*)(wt2 + ka);
    const v2f b3 = *(const v2f*)(wt3 + ka);
    acc0 = __builtin_amdgcn_wmma_f32_16x16x4_f32(false, a, false, b0, (short)0, acc0, false, false);
    acc1 = __builtin_amdgcn_wmma_f32_16x16x4_f32(false, a, false, b1, (short)0, acc1, false, false);
    acc2 = __builtin_amdgcn_wmma_f32_16x16x4_f32(false, a, false, b2, (short)0, acc2, false, false);
    acc3 = __builtin_amdgcn_wmma_f32_16x16x4_f32(false, a, false, b3, (short)0, acc3, false, false);
  }

  // C/D 16x16 f32: VGPR r -> row r (lanes 0-15) / row r+8 (lanes 16-31), col = lane&15
  v8f accs[4] = {acc0, acc1, acc2, acc3};
#pragma unroll
  for (int t = 0; t < 4; ++t) {
    const int col = nt0 + t * 16 + l;
    const float bv = bias ? bias[col] : 0.0f;
#pragma unroll
    for (int r = 0; r < 8; ++r) {
      const int row = mt + r + half * 8;
      H[(size_t)row * N + col] = accs[t][r] + bv;
    }
  }
}

// ----------------------------------------- edge aggregation (scatter-add)
// One wave per edge; 32 lanes x float4 = 128 channels. Atomics resolve in L2
// (feature buffer is 25.6 MB << 192 MB L2).
__global__ void aggregate_edges(const float* __restrict__ H, const int* __restrict__ src,
                                const int* __restrict__ dst, const float* __restrict__ dinv,
                                float* __restrict__ out, int E) {
  const int e = blockIdx.x * (blockDim.x >> 5) + (threadIdx.x >> 5);
  if (e >= E) return;
  const int lane = threadIdx.x & 31;
  const int s = src[e];
  const int d = dst[e];
  const float nrm = dinv[s] * dinv[d];
  const float4 v = ((const float4*)(H + (size_t)s * HID))[lane];
  float* o = out + (size_t)d * HID + lane * 4;
  atomicAdd(o + 0, v.x * nrm);
  atomicAdd(o + 1, v.y * nrm);
  atomicAdd(o + 2, v.z * nrm);
  atomicAdd(o + 3, v.w * nrm);
}

// self-loop contribution (h[i]*dinv[i]^2) + bias + optional ReLU, float4/thread
__global__ void selfloop_bias_act(const float* __restrict__ H, const float* __restrict__ dinv,
                                  const float* __restrict__ bias, float* __restrict__ out,
                                  int N, int relu) {
  const int i = blockIdx.x * blockDim.x + threadIdx.x;
  if (i >= N * (HID / 4)) return;
  const int node = i >> 5;          // HID/4 == 32
  const int c4   = i & 31;
  const float di  = dinv[node];
  const float nrm = di * di;
  const float4 hv = ((const float4*)(H + (size_t)node * HID))[c4];
  const float4 bb = ((const float4*)bias)[c4];
  float4* op = (float4*)(out + (size_t)node * HID) + c4;
  float4 v = *op;
  v.x += hv.x * nrm + bb.x;
  v.y += hv.y * nrm + bb.y;
  v.z += hv.z * nrm + bb.z;
  v.w += hv.w * nrm + bb.w;
  if (relu) {
    v.x = fmaxf(v.x, 0.0f); v.y = fmaxf(v.y, 0.0f);
    v.z = fmaxf(v.z, 0.0f); v.w = fmaxf(v.w, 0.0f);
  }
  *op = v;
}

// ---------------------------------------------------------------- pooling
__global__ void pool_accum(const float* __restrict__ x, const int* __restrict__ batch,
                           float* __restrict__ sums, float* __restrict__ counts, int N) {
  const int i = blockIdx.x * blockDim.x + threadIdx.x;
  if (i >= N * (HID / 4)) return;
  const int node = i >> 5;
  const int c4   = i & 31;
  const int g    = batch[node];
  const float4 v = ((const float4*)(x + (size_t)node * HID))[c4];
  float* s = sums + (size_t)g * HID + c4 * 4;
  atomicAdd(s + 0, v.x);
  atomicAdd(s + 1, v.y);
  atomicAdd(s + 2, v.z);
  atomicAdd(s + 3, v.w);
  if (c4 == 0) atomicAdd(counts + g, 1.0f);
}

__global__ void pool_divide(const float* __restrict__ sums, const float* __restrict__ counts,
                            float* __restrict__ pooled) {
  const int i = blockIdx.x * blockDim.x + threadIdx.x;
  if (i >= N_GRAPHS * HID) return;
  const int g = i / HID;
  pooled[i] = sums[i] / fmaxf(counts[g], 1.0f);
}

// ------------------------------------------------------------ L2 distance
__global__ void pairwise_distance(const float* __restrict__ e1, const float* __restrict__ e2,
                                  float* __restrict__ out) {
  const int g = blockIdx.x * blockDim.x + threadIdx.x;
  if (g >= N_GRAPHS) return;
  float acc = 0.0f;
#pragma unroll
  for (int c = 0; c < DOUT; ++c) {
    const float d = e1[(size_t)g * DOUT + c] - e2[(size_t)g * DOUT + c] + EPS_PD;
    acc += d * d;
  }
  out[g] = sqrtf(acc);
}

// ---------------------------------------------------------------- driver
static void run_branch(const float* x, const int* edge_index, const int* batch,
                       const float* wt1, const float* b1, const float* wt2, const float* b2,
                       const float* wt3, const float* b3, const float* wtlin, const float* blin,
                       float* dinv, float* bufA, float* bufB, float* sums, float* counts,
                       float* pooled, float* emb, int N, int E, hipStream_t stream) {
  const int* src = edge_index;       // edge_index[0,:]
  const int* dst = edge_index + E;   // edge_index[1,:]

  // degree (self-loop contributes 1) -> dinv
  fill_f32<<<ceil_div(N, 256), 256, 0, stream>>>(dinv, 1.0f, N);
  degree_accum<<<ceil_div(E, 256), 256, 0, stream>>>(dst, dinv, E);
  rsqrt_inplace<<<ceil_div(N, 256), 256, 0, stream>>>(dinv, N);

  const dim3 gblk(32, HID / 64);     // 2 waves per block, 16x64 strip each
  const dim3 ggrd(N / 16, 1);
  const int feat4 = N * (HID / 4);

  const float* xin = x;
  const float* Ws[3] = {wt1, wt2, wt3};
  const float* bs[3] = {b1, b2, b3};
  for (int layer = 0; layer < 3; ++layer) {
    // h = x @ W  (fp32 WMMA, W pre-transposed)
    wmma_gemm_f32<<<ggrd, gblk, 0, stream>>>(xin, Ws[layer], nullptr, bufA, N, HID, HID);
    // out = scatter_add(h[src] * norm) + h*dinv^2 + b, relu on first two layers
    fill_f32<<<ceil_div(N * HID, 256), 256, 0, stream>>>(bufB, 0.0f, N * HID);
    aggregate_edges<<<ceil_div(E, 8), 256, 0, stream>>>(bufA, src, dst, dinv, bufB, E);
    selfloop_bias_act<<<ceil_div(feat4, 256), 256, 0, stream>>>(bufA, dinv, bs[layer], bufB,
                                                               N, layer < 2 ? 1 : 0);
    xin = bufB;
  }

  // mean pool
  fill_f32<<<ceil_div(N_GRAPHS * HID, 256), 256, 0, stream>>>(sums, 0.0f, N_GRAPHS * HID);
  fill_f32<<<ceil_div(N_GRAPHS, 256), 256, 0, stream>>>(counts, 0.0f, N_GRAPHS);
  pool_accum<<<ceil_div(feat4, 256), 256, 0, stream>>>(bufB, batch, sums, counts, N);
  pool_divide<<<ceil_div(N_GRAPHS * HID, 256), 256, 0, stream>>>(sums, counts, pooled);

  // final linear (WMMA, 512x128 @ 128x64 + bias), one 16x64 strip per wave
  wmma_gemm_f32<<<dim3(N_GRAPHS / 16, 1), dim3(32, 1), 0, stream>>>(pooled, wtlin, blin, emb,
                                                                    N_GRAPHS, DOUT, HID);
}

extern "C" void kernel_launch(void* const* d_in, const int* in_sizes, int n_in,
                              void* d_out, int out_size, void* d_ws, size_t ws_size,
                              hipStream_t stream) {
  const float* x1     = (const float*)d_in[0];
  const int*   ei1    = (const int*)d_in[1];
  const int*   batch1 = (const int*)d_in[2];
  const float* x2     = (const float*)d_in[3];
  const int*   ei2    = (const int*)d_in[4];
  const int*   batch2 = (const int*)d_in[5];
  const float* W1   = (const float*)d_in[6];
  const float* b1   = (const float*)d_in[7];
  const float* W2   = (const float*)d_in[8];
  const float* b2   = (const float*)d_in[9];
  const float* W3   = (const float*)d_in[10];
  const float* b3   = (const float*)d_in[11];
  const float* Wlin = (const float*)d_in[12];
  const float* blin = (const float*)d_in[13];

  const int N = in_sizes[0] / HID;   // 50000 (multiple of 16)
  const int E = in_sizes[1] / 2;     // 1600000

  // carve workspace
  char* ws = (char*)d_ws;
  auto take = [&](size_t bytes) {
    char* p = ws;
    ws += (bytes + 255) & ~(size_t)255;
    return (float*)p;
  };
  float* dinv   = take((size_t)N * 4);
  float* bufA   = take((size_t)N * HID * 4);
  float* bufB   = take((size_t)N * HID * 4);
  float* sums   = take((size_t)N_GRAPHS * HID * 4);
  float* counts = take((size_t)N_GRAPHS * 4);
  float* pooled = take((size_t)N_GRAPHS * HID * 4);
  float* emb1   = take((size_t)N_GRAPHS * DOUT * 4);
  float* emb2   = take((size_t)N_GRAPHS * DOUT * 4);
  float* wt1    = take((size_t)HID * HID * 4);
  float* wt2    = take((size_t)HID * HID * 4);
  float* wt3    = take((size_t)HID * HID * 4);
  float* wtlin  = take((size_t)HID * DOUT * 4);
  (void)ws_size; (void)n_in; (void)out_size;

  // pre-transpose weights once (shared by both branches)
  transpose_w<<<ceil_div(HID * HID, 256), 256, 0, stream>>>(W1, wt1, HID, HID);
  transpose_w<<<ceil_div(HID * HID, 256), 256, 0, stream>>>(W2, wt2, HID, HID);
  transpose_w<<<ceil_div(HID * HID, 256), 256, 0, stream>>>(W3, wt3, HID, HID);
  transpose_w<<<ceil_div(HID * DOUT, 256), 256, 0, stream>>>(Wlin, wtlin, HID, DOUT);

  run_branch(x1, ei1, batch1, wt1, b1, wt2, b2, wt3, b3, wtlin, blin,
             dinv, bufA, bufB, sums, counts, pooled, emb1, N, E, stream);
  run_branch(x2, ei2, batch2, wt1, b1, wt2, b2, wt3, b3, wtlin, blin,
             dinv, bufA, bufB, sums, counts, pooled, emb2, N, E, stream);

  pairwise_distance<<<ceil_div(N_GRAPHS, 256), 256, 0, stream>>>(emb1, emb2, (float*)d_out);
}